// TopKRouter_50646254355291
// MI455X (gfx1250) — compile-verified
//
#include <hip/hip_runtime.h>
#include <hip/hip_bf16.h>

typedef __attribute__((ext_vector_type(2))) float v2f;
typedef __attribute__((ext_vector_type(4))) float v4f;
typedef __attribute__((ext_vector_type(8))) float v8f;

constexpr int TOKENS    = 32768;
constexpr int HIDDEN    = 2048;
constexpr int EXPERTS   = 64;
constexpr int TOPK      = 8;
constexpr int WG_TOKENS = 128;  // 4 waves x 32 tokens
constexpr int LDSSTRIDE = 65;   // padded row -> conflict-free column scans

__global__ __launch_bounds__(128) void router_fused(
    const float* __restrict__ H,
    const float* __restrict__ W,
    const float* __restrict__ Bb,
    float* __restrict__ scores,
    int*   __restrict__ top_idx)
{
    __shared__ float lds[WG_TOKENS * LDSSTRIDE];

    const int tid    = threadIdx.x;
    const int wave   = tid >> 5;
    const int lane   = tid & 31;
    const int laneLo = lane & 15;
    const int laneHi = lane >> 4;               // 0: K+0..1  1: K+2..3
    const int wgTok0   = blockIdx.x * WG_TOKENS;
    const int waveTok0 = wgTok0 + wave * 32;    // 32 tokens per wave (two 16-row A tiles)

    // ---------- Phase 1: router GEMM (32 tokens x 64 experts per wave) ----------
    // V_WMMA_F32_16X16X4_F32: A 16x4 (2 VGPR), B 4x16 (2 VGPR), C 16x16 (8 VGPR)
    v8f acc0 = {}, acc1 = {}, acc2 = {}, acc3 = {};   // tokens +0..15, expert tiles 0..3
    v8f acc4 = {}, acc5 = {}, acc6 = {}, acc7 = {};   // tokens +16..31, expert tiles 0..3

    const float* arowA = H + (size_t)(waveTok0 +  0 + laneLo) * HIDDEN + 2 * laneHi;
    const float* arowB = H + (size_t)(waveTok0 + 16 + laneLo) * HIDDEN + 2 * laneHi;
    const float* wr0   = W + (size_t)( 0 + laneLo) * HIDDEN + 2 * laneHi;
    const float* wr1   = W + (size_t)(16 + laneLo) * HIDDEN + 2 * laneHi;
    const float* wr2   = W + (size_t)(32 + laneLo) * HIDDEN + 2 * laneHi;
    const float* wr3   = W + (size_t)(48 + laneLo) * HIDDEN + 2 * laneHi;

    for (int kb = 0; kb < HIDDEN; kb += 32) {
        // WGP-scope prefetch (scope 0 = fill all cache levels) for the streamed
        // activation rows, 1 KB (8 iterations) ahead. The clang builtin can only
        // reach SE/SYS scope, which skips the WGP cache, so use inline asm.
        asm volatile("global_prefetch_b8 %0, off offset:1024"
                     :: "v"(arowA + kb) : "memory");
        asm volatile("global_prefetch_b8 %0, off offset:1024"
                     :: "v"(arowB + kb) : "memory");
#pragma unroll
        for (int k = kb; k < kb + 32; k += 4) {
            v2f a0 = *(const v2f*)(arowA + k);
            v2f a1 = *(const v2f*)(arowB + k);
            v2f b0 = *(const v2f*)(wr0 + k);
            v2f b1 = *(const v2f*)(wr1 + k);
            v2f b2 = *(const v2f*)(wr2 + k);
            v2f b3 = *(const v2f*)(wr3 + k);
            acc0 = __builtin_amdgcn_wmma_f32_16x16x4_f32(false, a0, false, b0, (short)0, acc0, false, false);
            acc1 = __builtin_amdgcn_wmma_f32_16x16x4_f32(false, a0, false, b1, (short)0, acc1, false, false);
            acc2 = __builtin_amdgcn_wmma_f32_16x16x4_f32(false, a0, false, b2, (short)0, acc2, false, false);
            acc3 = __builtin_amdgcn_wmma_f32_16x16x4_f32(false, a0, false, b3, (short)0, acc3, false, false);
            acc4 = __builtin_amdgcn_wmma_f32_16x16x4_f32(false, a1, false, b0, (short)0, acc4, false, false);
            acc5 = __builtin_amdgcn_wmma_f32_16x16x4_f32(false, a1, false, b1, (short)0, acc5, false, false);
            acc6 = __builtin_amdgcn_wmma_f32_16x16x4_f32(false, a1, false, b2, (short)0, acc6, false, false);
            acc7 = __builtin_amdgcn_wmma_f32_16x16x4_f32(false, a1, false, b3, (short)0, acc7, false, false);
        }
    }

    // bias add: lane's column is expert n*16 + laneLo for every row of acc n
    const float bb0 = Bb[ 0 + laneLo], bb1 = Bb[16 + laneLo];
    const float bb2 = Bb[32 + laneLo], bb3 = Bb[48 + laneLo];
#pragma unroll
    for (int r = 0; r < 8; ++r) {
        acc0[r] += bb0; acc1[r] += bb1; acc2[r] += bb2; acc3[r] += bb3;
        acc4[r] += bb0; acc5[r] += bb1; acc6[r] += bb2; acc7[r] += bb3;
    }

    // spill tiles to LDS: C layout VGPR r = row r (lanes 0-15) / row r+8 (lanes 16-31), col = laneLo
#pragma unroll
    for (int r = 0; r < 8; ++r) {
        const int ltA = wave * 32 + r + 8 * laneHi;        // tokens +0..15
        const int ltB = ltA + 16;                          // tokens +16..31
        lds[ltA * LDSSTRIDE +  0 + laneLo] = acc0[r];
        lds[ltA * LDSSTRIDE + 16 + laneLo] = acc1[r];
        lds[ltA * LDSSTRIDE + 32 + laneLo] = acc2[r];
        lds[ltA * LDSSTRIDE + 48 + laneLo] = acc3[r];
        lds[ltB * LDSSTRIDE +  0 + laneLo] = acc4[r];
        lds[ltB * LDSSTRIDE + 16 + laneLo] = acc5[r];
        lds[ltB * LDSSTRIDE + 32 + laneLo] = acc6[r];
        lds[ltB * LDSSTRIDE + 48 + laneLo] = acc7[r];
    }
    __syncthreads();

    // ---------- Phase 2: per-token top-8 + softmax + scatter (all 128 threads) ----------
    {
        const float* row = &lds[tid * LDSSTRIDE];
        unsigned long long used = 0ull;
        float vals[TOPK]; int idxs[TOPK];
#pragma unroll
        for (int j = 0; j < TOPK; ++j) {
            float best = -3.402823e38f; int bi = 0;
            for (int e = 0; e < EXPERTS; ++e) {
                float v = row[e];
                bool ok = (((used >> e) & 1ull) == 0ull) && (v > best);
                best = ok ? v : best;
                bi   = ok ? e : bi;
            }
            used |= (1ull << bi);
            vals[j] = best; idxs[j] = bi;
        }
        float p[TOPK]; float s = 0.f;           // vals[0] is the max
#pragma unroll
        for (int j = 0; j < TOPK; ++j) { p[j] = __expf(vals[j] - vals[0]); s += p[j]; }
        const float inv = 1.0f / s;

        const int token = wgTok0 + tid;
        float* srow = scores + (size_t)token * EXPERTS;
        v4f z = {};
#pragma unroll
        for (int e = 0; e < EXPERTS; e += 4) *(v4f*)(srow + e) = z;
#pragma unroll
        for (int j = 0; j < TOPK; ++j) {
            srow[idxs[j]] = p[j] * inv;
            top_idx[(size_t)token * TOPK + j] = idxs[j];
        }
    }
}

extern "C" void kernel_launch(void* const* d_in, const int* in_sizes, int n_in,
                              void* d_out, int out_size, void* d_ws, size_t ws_size,
                              hipStream_t stream) {
    const float* H  = (const float*)d_in[0];   // [32768, 2048]
    const float* W  = (const float*)d_in[1];   // [64, 2048]
    const float* Bb = (const float*)d_in[2];   // [64]
    float* scores  = (float*)d_out;                                    // [32768, 64]
    int*   top_idx = (int*)((float*)d_out + (size_t)TOKENS * EXPERTS); // [32768, 8]
    router_fused<<<dim3(TOKENS / WG_TOKENS), 128, 0, stream>>>(H, W, Bb, scores, top_idx);
}